// UpsamplingLayer_16003048144965
// MI455X (gfx1250) — compile-verified
//
#include <hip/hip_runtime.h>
#include <cstdint>
#include <cstddef>

// 2x upsample (zero-interleave + 4x4 [1,3,3,1]^2 blur, gain=4) of
// x:(16,128,128,128) f32 -> out:(16,128,256,256) f32.
// Separable polyphase: per-dim taps {0.25, 0.75} per output phase.
//
// Bandwidth-bound (~640 MiB traffic, ~1.5 flop/byte): async global->LDS
// tile staging (CDNA5 ASYNCcnt path, mostly 128-bit transfers) + 128-bit
// non-temporal coalesced stores.

#define TI        32     // input tile edge  -> 64x64 output tile
#define HALO_ROWS 34     // TI + 1 halo row on each side
#define LDS_W     40     // padded LDS row stride (floats); interior at col 4 -> 16B aligned
#define COL_OFF   4      // smem col s corresponds to x col (j0 + s - COL_OFF)
#define IN_W      128
#define OUT_W     256

typedef float v4f __attribute__((ext_vector_type(4)));

__global__ __launch_bounds__(256)
void upsample2x_blur_kernel(const float* __restrict__ x, float* __restrict__ out)
{
    __shared__ float smem[HALO_ROWS * LDS_W];

    const int plane = blockIdx.y;            // fused n*c plane: 0..2047
    const int tx = blockIdx.x & 3;           // tile col 0..3
    const int ty = blockIdx.x >> 2;          // tile row 0..3
    const int i0 = ty * TI;
    const int j0 = tx * TI;

    const float* xp = x   + (size_t)plane * (IN_W * IN_W);
    float*       op = out + (size_t)plane * (OUT_W * OUT_W);

    const int t = threadIdx.x;

    // ---- Stage tile interior: 34 rows x 8 quads via async b128 global->LDS ----
    // LDS dest byte offset = r*160 + 16 + q*16  -> 16B aligned.
    for (int u = t; u < HALO_ROWS * 8; u += 256) {
        const int r = u >> 3;                 // 0..33
        const int q = u & 7;                  // 0..7
        const int gi = i0 + r - 1;
        float* lp = &smem[r * LDS_W + COL_OFF + q * 4];
        if ((unsigned)gi < (unsigned)IN_W) {
            const float* gp = xp + gi * IN_W + j0 + q * 4;
            asm volatile("global_load_async_to_lds_b128 %0, %1, off"
                         :: "v"((uint32_t)(uintptr_t)lp), "v"(gp)
                         : "memory");
        } else {
            *(v4f*)lp = (v4f){0.f, 0.f, 0.f, 0.f};   // zero-pad out-of-plane rows
        }
    }
    // ---- Stage halo columns: 34 rows x 2 cols via async b32 ----
    if (t < HALO_ROWS * 2) {
        const int r    = t >> 1;
        const int side = t & 1;
        const int gi = i0 + r - 1;
        const int gj = side ? (j0 + TI) : (j0 - 1);
        float* lp = &smem[r * LDS_W + (side ? (COL_OFF + TI) : (COL_OFF - 1))];
        if ((unsigned)gi < (unsigned)IN_W && (unsigned)gj < (unsigned)IN_W) {
            const float* gp = xp + gi * IN_W + gj;
            asm volatile("global_load_async_to_lds_b32 %0, %1, off"
                         :: "v"((uint32_t)(uintptr_t)lp), "v"(gp)
                         : "memory");
        } else {
            *lp = 0.0f;                        // zero-pad plane borders
        }
    }
    // Each wave drains its own async transfers; barrier (waits DScnt too)
    // makes the tile visible to all 8 waves.
    asm volatile("s_wait_asynccnt 0x0" ::: "memory");
    __syncthreads();

    // ---- Compute: each thread -> 2 output rows x 8 output cols ----
    const int qr = t >> 3;            // input row within tile: 0..31
    const int qc = (t & 7) << 2;      // input col base within tile: 0,4,...,28

    const float A = 0.25f, B = 0.75f;

    // Horizontal pass for 3 input rows (i-1, i, i+1), 4 input cols each.
    // smem[(qr+rr)*LDS_W + qc + COL_OFF - 1 + k] == x[i0+qr+rr-1][j0+qc+k-1]
    float he[3][4], ho[3][4];
#pragma unroll
    for (int rr = 0; rr < 3; ++rr) {
        const float* row = &smem[(qr + rr) * LDS_W + qc + (COL_OFF - 1)];
#pragma unroll
        for (int k = 0; k < 4; ++k) {
            const float xm = row[k];        // x[j-1]
            const float xc = row[k + 1];    // x[j]
            const float xq = row[k + 2];    // x[j+1]
            he[rr][k] = fmaf(A, xm, B * xc);   // even out col 2j
            ho[rr][k] = fmaf(B, xc, A * xq);   // odd  out col 2j+1
        }
    }

    // Vertical pass: even out row 2i uses rows (i-1,i); odd row 2i+1 uses (i,i+1).
    float oe[8], oo[8];
#pragma unroll
    for (int k = 0; k < 4; ++k) {
        oe[2 * k]     = fmaf(A, he[0][k], B * he[1][k]);
        oe[2 * k + 1] = fmaf(A, ho[0][k], B * ho[1][k]);
        oo[2 * k]     = fmaf(B, he[1][k], A * he[2][k]);
        oo[2 * k + 1] = fmaf(B, ho[1][k], A * ho[2][k]);
    }

    const int orow = 2 * (i0 + qr);
    const int ocol = 2 * (j0 + qc);          // multiple of 8 -> 16B aligned
    float* pe = op + (size_t)orow * OUT_W + ocol;
    float* po = pe + OUT_W;

    // Write-once streaming output: non-temporal 128-bit stores.
    v4f ve0 = { oe[0], oe[1], oe[2], oe[3] };
    v4f ve1 = { oe[4], oe[5], oe[6], oe[7] };
    v4f vo0 = { oo[0], oo[1], oo[2], oo[3] };
    v4f vo1 = { oo[4], oo[5], oo[6], oo[7] };
    __builtin_nontemporal_store(ve0, (v4f*)(pe));
    __builtin_nontemporal_store(ve1, (v4f*)(pe + 4));
    __builtin_nontemporal_store(vo0, (v4f*)(po));
    __builtin_nontemporal_store(vo1, (v4f*)(po + 4));
}

extern "C" void kernel_launch(void* const* d_in, const int* in_sizes, int n_in,
                              void* d_out, int out_size, void* d_ws, size_t ws_size,
                              hipStream_t stream) {
    (void)in_sizes; (void)n_in; (void)out_size; (void)d_ws; (void)ws_size;
    const float* x = (const float*)d_in[0];
    // d_in[1] is the 4x4 kernel; it is the fixed [1,3,3,1] binomial with gain 4,
    // whose separable per-phase taps {0.25, 0.75} are hardcoded above.
    float* out = (float*)d_out;

    // 2048 planes (N*C), 16 tiles (4x4) of 32x32 input per plane.
    dim3 grid(16, 16 * 128);
    upsample2x_blur_kernel<<<grid, 256, 0, stream>>>(x, out);
}